// SingleGatLayer_36515811951252
// MI455X (gfx1250) — compile-verified
//
#include <hip/hip_runtime.h>
#include <math.h>

typedef __attribute__((ext_vector_type(2))) float v2f;
typedef __attribute__((ext_vector_type(4))) float v4f;
typedef __attribute__((ext_vector_type(8))) float v8f;

#define B_      8
#define N_      4096
#define K_      26
#define KP_     (K_ / 2)        // 13 k-pairs
#define D_      64
#define CHUNKS  32              // n-chunks per batch -> 256 blocks in kernel 1
#define NPB     (N_ / CHUNKS)   // 128 n per block
#define WAVES   8
#define THREADS 256
#define ELEMS   (K_ * D_)       // 1664 outputs per b
#define OUTTOT  (B_ * ELEMS)    // 13312 total outputs
#define GROUPS  (ELEMS / 16)    // 104 column-groups of 16 per b

// ---------------------------------------------------------------------------
// Kernel 1: stream neighbors once (bandwidth-bound, 218 MB) with non-temporal
// global_load_b128: each half-wave owns one k of a k-pair, each lane a float4
// of the D axis. Accumulate all 26 k in 52 VGPRs per lane, stage to LDS, then
// reduce the 8 wave-copies with V_WMMA_F32_16X16X4_F32 (all-ones A = row-sum
// reducer), writing per-chunk partials to workspace.
// ---------------------------------------------------------------------------
__global__ __launch_bounds__(THREADS)
void gat_partial_kernel(const float* __restrict__ targets,
                        const float* __restrict__ neighbors,
                        const float* __restrict__ a,
                        const float* __restrict__ kernel_target,
                        const float* __restrict__ bias_target,
                        const float* __restrict__ kernel_neighbor,
                        const float* __restrict__ bias_neighbor,
                        float* __restrict__ partial) {
    __shared__ float lds[WAVES * ELEMS];   // 53,248 B (WGP has 320 KB)

    const int tid   = threadIdx.x;
    const int lane  = tid & 31;
    const int wave  = tid >> 5;
    const int half  = lane >> 4;           // 0: k even, 1: k odd of the pair
    const int b     = blockIdx.x / CHUNKS;
    const int chunk = blockIdx.x % CHUNKS;
    const int nbase = chunk * NPB;

    const int dq = (lane & 15) * 4;        // this lane's float4 of the D axis
    // Note the reference's swapped naming: targets get kernel_neighbor/bias_neighbor.
    const v4f kn4 = *(const v4f*)(kernel_neighbor + dq);
    const v4f bn4 = *(const v4f*)(bias_neighbor   + dq);
    const v4f kt4 = *(const v4f*)(kernel_target   + dq);
    const v4f bt4 = *(const v4f*)(bias_target     + dq);

    v4f acc[KP_];
    #pragma unroll
    for (int kp = 0; kp < KP_; ++kp) acc[kp] = (v4f){0.f, 0.f, 0.f, 0.f};

    for (int i = 0; i < NPB / WAVES; ++i) {
        const int n = nbase + wave + i * WAVES;      // each wave owns distinct n's
        const size_t row = (size_t)b * N_ + (size_t)n;

        const v4f tv = *(const v4f*)(targets + row * D_ + dq);
        const v4f tt = tv * kn4 + bn4;

        const float* arow = a + row * K_;
        const float* nrow = neighbors + row * (size_t)ELEMS;

        #pragma unroll
        for (int kp = 0; kp < KP_; ++kp) {
            // lane's 16B slice of the k-pair block: 512 B coalesced per wave,
            // non-temporal (read-once stream, don't thrash the 192 MB L2)
            const v4f nb = __builtin_nontemporal_load(
                (const v4f*)(nrow + kp * (2 * D_) + lane * 4));
            const v2f av2 = *(const v2f*)(arow + 2 * kp);
            const float av = half ? av2.y : av2.x;   // this half-wave's a[k]
            const v4f nn = nb * kt4 + bt4;
            acc[kp] += (tt * nn) * av;
        }
    }

    // Stage per-wave accumulators to LDS in canonical [wave][k*64 + d] layout
    #pragma unroll
    for (int kp = 0; kp < KP_; ++kp) {
        const int k = 2 * kp + half;
        *(v4f*)&lds[wave * ELEMS + k * D_ + dq] = acc[kp];
    }
    __syncthreads();

    // Cross-wave reduction: sum the 8 wave copies per output element with two
    // chained WMMAs (4 rows each). A = all-ones -> D[:,col] = sum of B rows,
    // replicated across all lanes, so the result is layout-robust.
    const v2f ones = {1.f, 1.f};
    const int col  = lane & 15;
    const int wsel = (lane < 16) ? 0 : 2;   // B VGPR0 half-wave row split
    for (int g = wave; g < GROUPS; g += WAVES) {      // 104/8 = 13 iters, uniform
        const int cbase = g * 16;
        v8f c = {0.f, 0.f, 0.f, 0.f, 0.f, 0.f, 0.f, 0.f};
        v2f b0, b1;
        b0.x = lds[(wsel + 0) * ELEMS + cbase + col]; // waves 0..3
        b0.y = lds[(wsel + 1) * ELEMS + cbase + col];
        b1.x = lds[(wsel + 4) * ELEMS + cbase + col]; // waves 4..7
        b1.y = lds[(wsel + 5) * ELEMS + cbase + col];
        c = __builtin_amdgcn_wmma_f32_16x16x4_f32(false, ones, false, b0,
                                                  (short)0, c, false, false);
        c = __builtin_amdgcn_wmma_f32_16x16x4_f32(false, ones, false, b1,
                                                  (short)0, c, false, false);
        if (lane < 16) {
            partial[(size_t)chunk * OUTTOT + (size_t)b * ELEMS + cbase + col] = c[0];
        }
    }
}

// ---------------------------------------------------------------------------
// Kernel 2: reduce the 32 chunk-partials per output (8 chained WMMAs with the
// ones-A reducer), apply sigmoid, store final (1,B,K,D) result.
// ---------------------------------------------------------------------------
__global__ __launch_bounds__(THREADS)
void gat_reduce_kernel(const float* __restrict__ partial,
                       float* __restrict__ out) {
    const int lane  = threadIdx.x & 31;
    const int wave  = threadIdx.x >> 5;
    const int g     = blockIdx.x * WAVES + wave;   // 0..831 column groups
    const int cbase = g * 16;
    const int col   = lane & 15;
    const int rsel  = (lane < 16) ? 0 : 2;

    const v2f ones = {1.f, 1.f};
    v8f c = {0.f, 0.f, 0.f, 0.f, 0.f, 0.f, 0.f, 0.f};

    #pragma unroll
    for (int r = 0; r < CHUNKS; r += 4) {
        v2f bm;
        bm.x = partial[(size_t)(r + rsel + 0) * OUTTOT + cbase + col];
        bm.y = partial[(size_t)(r + rsel + 1) * OUTTOT + cbase + col];
        c = __builtin_amdgcn_wmma_f32_16x16x4_f32(false, ones, false, bm,
                                                  (short)0, c, false, false);
    }

    if (lane < 16) {
        const float w = c[0];
        out[cbase + col] = 1.0f / (1.0f + __expf(-w));
    }
}

// ---------------------------------------------------------------------------
// Launch. d_in order (setup_inputs): targets, neighbors, a, kernel_target,
// bias_target, kernel_neighbor, bias_neighbor — all float32.
// Workspace need: CHUNKS * OUTTOT * 4 = 1,703,936 bytes.
// ---------------------------------------------------------------------------
extern "C" void kernel_launch(void* const* d_in, const int* in_sizes, int n_in,
                              void* d_out, int out_size, void* d_ws, size_t ws_size,
                              hipStream_t stream) {
    const float* targets         = (const float*)d_in[0];
    const float* neighbors       = (const float*)d_in[1];
    const float* a               = (const float*)d_in[2];
    const float* kernel_target   = (const float*)d_in[3];
    const float* bias_target     = (const float*)d_in[4];
    const float* kernel_neighbor = (const float*)d_in[5];
    const float* bias_neighbor   = (const float*)d_in[6];
    float* partial = (float*)d_ws;
    float* out     = (float*)d_out;

    gat_partial_kernel<<<B_ * CHUNKS, THREADS, 0, stream>>>(
        targets, neighbors, a, kernel_target, bias_target,
        kernel_neighbor, bias_neighbor, partial);

    gat_reduce_kernel<<<(OUTTOT / 16) / WAVES, THREADS, 0, stream>>>(partial, out);
}